// MultiLinear_1580547967571
// MI455X (gfx1250) — compile-verified
//
#include <hip/hip_runtime.h>

typedef float v2f __attribute__((ext_vector_type(2)));
typedef float v8f __attribute__((ext_vector_type(8)));

#define N_SAMPLES 2048
#define D_DIM     512
#define H_DIM     512
#define NH        16
#define ROW_TILES_MAX (N_SAMPLES / 16)   // 128 (worst case: all rows one head)
#define LDS_STRIDE 516                   // 512 + 4 pad -> conflict-free WMMA frag reads

// ---------------- Phase 1: bucket rows by head ----------------

__global__ void zero_counts_kernel(int* __restrict__ counts) {
    if (threadIdx.x < NH) counts[threadIdx.x] = 0;
}

__global__ void bucket_kernel(const long long* __restrict__ idx,
                              int* __restrict__ counts,
                              int* __restrict__ rowlist) {
    int n = blockIdx.x * blockDim.x + threadIdx.x;
    if (n >= N_SAMPLES) return;
    int h = (int)idx[n];
    int pos = atomicAdd(&counts[h], 1);
    rowlist[h * N_SAMPLES + pos] = n;
}

// ---------------- Phase 2: grouped GEMM via V_WMMA_F32_16X16X4_F32 ----------------
// grid.x = NH * ROW_TILES_MAX (head-major), grid.y = 4 col groups (8 waves x 16 cols each)
// block  = 256 threads = 8 wave32s; wave w owns col tile (blockIdx.y*8 + w).

__global__ __launch_bounds__(256)
void grouped_gemm_kernel(const float* __restrict__ x,       // (N, D)
                         const float* __restrict__ w,       // (NH, D, H)
                         const float* __restrict__ bias,    // (NH, H)
                         const int*  __restrict__ counts,   // (NH)
                         const int*  __restrict__ rowlist,  // (NH, N)
                         float* __restrict__ out) {         // (N, H)
    const int h  = blockIdx.x / ROW_TILES_MAX;
    const int rt = blockIdx.x % ROW_TILES_MAX;
    const int cnt = counts[h];
    if (rt * 16 >= cnt) return;               // empty row-tile for this head

    __shared__ float ldsA[16 * LDS_STRIDE];   // 16 gathered A rows, padded stride
    __shared__ int   rows[16];

    const int tid = threadIdx.x;
    if (tid < 16) {
        int p = rt * 16 + tid;
        rows[tid] = (p < cnt) ? rowlist[h * N_SAMPLES + p] : -1;
    }
    __syncthreads();

    // Cooperative gather of 16 rows x 512 f32 into LDS (float4 per thread x 8).
    // 16 rows * 128 float4 = 2048 float4 slots / 256 threads = 8 each.
    #pragma unroll
    for (int e = 0; e < 8; ++e) {
        int lin = tid + e * 256;
        int r   = lin >> 7;          // row within tile 0..15
        int q   = lin & 127;         // float4 index within row 0..127
        int row = rows[r];
        float4 v = make_float4(0.f, 0.f, 0.f, 0.f);
        if (row >= 0) v = ((const float4*)(x + (size_t)row * D_DIM))[q];
        *(float4*)&ldsA[r * LDS_STRIDE + q * 4] = v;
    }
    __syncthreads();

    const int wave = tid >> 5;
    const int lane = tid & 31;
    const int m16  = lane & 15;      // M (for A) / N (for B,C) sub-index
    const int hi   = lane >> 4;      // 0: K pair {0,1}; 1: K pair {2,3}

    const int ct    = blockIdx.y * 8 + wave;   // col tile 0..31
    const int ncol  = ct * 16 + m16;           // this lane's output column
    const float* wh = w + (size_t)h * D_DIM * H_DIM;

    v8f c = {};
    // K loop: 512 / 4 = 128 WMMAs, unrolled x4 so several are in flight.
    #pragma unroll 4
    for (int k0 = 0; k0 < D_DIM; k0 += 4) {
        const int kk = k0 + 2 * hi;
        // A fragment: lane holds A[M=m16][kk], A[M=m16][kk+1] (ISA 16x4 f32 layout)
        v2f a = *(const v2f*)&ldsA[m16 * LDS_STRIDE + kk];
        // B fragment (mirrored): lane holds B[kk][ncol], B[kk+1][ncol]
        v2f b;
        b.x = wh[(size_t)(kk + 0) * H_DIM + ncol];
        b.y = wh[(size_t)(kk + 1) * H_DIM + ncol];
        // 8 args: (neg_a, A, neg_b, B, c_mod, C, reuse_a, reuse_b)
        c = __builtin_amdgcn_wmma_f32_16x16x4_f32(
                false, a, false, b, (short)0, c, false, false);
    }

    // Bias is per (head, col): same for every row of this tile.
    const float bv = bias[h * H_DIM + ncol];

    // C/D layout: VGPR r holds row M = r + 8*hi, column = ncol. Scatter to out.
    #pragma unroll
    for (int r = 0; r < 8; ++r) {
        int m   = r + 8 * hi;
        int row = rows[m];
        if (row >= 0) out[(size_t)row * H_DIM + ncol] = c[r] + bv;
    }
}

// ---------------- Launcher ----------------

extern "C" void kernel_launch(void* const* d_in, const int* in_sizes, int n_in,
                              void* d_out, int out_size, void* d_ws, size_t ws_size,
                              hipStream_t stream) {
    const float*     x    = (const float*)d_in[0];      // (2048, 512) f32
    const long long* idx  = (const long long*)d_in[1];  // (2048,) int64
    const float*     wgt  = (const float*)d_in[2];      // (16, 512, 512) f32
    const float*     bias = (const float*)d_in[3];      // (16, 512) f32
    float*           out  = (float*)d_out;              // (2048, 512) f32

    // Workspace layout: [0,64) counts (16 ints), [64, 64+16*2048*4) rowlist
    int* counts  = (int*)d_ws;
    int* rowlist = (int*)((char*)d_ws + 64);

    zero_counts_kernel<<<1, 32, 0, stream>>>(counts);
    bucket_kernel<<<(N_SAMPLES + 255) / 256, 256, 0, stream>>>(idx, counts, rowlist);

    dim3 grid(NH * ROW_TILES_MAX, H_DIM / (8 * 16));  // (2048, 4)
    grouped_gemm_kernel<<<grid, 256, 0, stream>>>(x, wgt, bias, counts, rowlist, out);
}